// S2ConvNet_68616397521198
// MI455X (gfx1250) — compile-verified
//
#include <hip/hip_runtime.h>
#include <math.h>
#include <stdint.h>

#define PI_F 3.14159265358979323846f
#define BN_EPS 1e-5f

typedef __attribute__((ext_vector_type(2))) float v2f;
typedef __attribute__((ext_vector_type(8))) float v8f;
typedef __attribute__((ext_vector_type(4))) unsigned int v4u;
typedef __attribute__((ext_vector_type(4))) int v4i;
typedef __attribute__((ext_vector_type(8))) int v8i;

// ---------------------------------------------------------------------------
// fp32 WMMA wrapper: D = A(16x4) * B(4x16) + C(16x16), wave32
// ---------------------------------------------------------------------------
__device__ __forceinline__ v8f wmma_f32(v2f a, v2f b, v8f c) {
  return __builtin_amdgcn_wmma_f32_16x16x4_f32(false, a, false, b, (short)0, c,
                                               false, false);
}

// ---------------------------------------------------------------------------
// TDM: issue one 2D fp32 tile load (32x32, row-major, stride ld elements)
// into LDS at byte offset ldsOff. tensor_dim = remaining extent so OOB reads
// return zero (hardware edge padding). pad_enable inserts 1 DWORD per 32
// DWORDs -> LDS row stride of 33 floats (bank-conflict-free column reads).
// ---------------------------------------------------------------------------
__device__ __forceinline__ void tdm_tile_f32(const float* g, int remRows,
                                             int remCols, int ld,
                                             unsigned ldsOff) {
  unsigned long long ga = (unsigned long long)(uintptr_t)g;
  unsigned d0 = (unsigned)(remCols > 0 ? remCols : 0);  // tensor_dim0 (cols)
  unsigned d1 = (unsigned)(remRows > 0 ? remRows : 0);  // tensor_dim1 (rows)
  unsigned long long st = (unsigned long long)(unsigned)ld;
  v4u g0;
  g0[0] = 1u;                                    // count=1 (valid user D#)
  g0[1] = ldsOff;                                // lds_addr (bytes)
  g0[2] = (unsigned)ga;                          // global_addr[31:0]
  g0[3] = (unsigned)((ga >> 32) & 0x01ffffffu)   // global_addr[56:32]
          | (2u << 30);                          // type = 2 ("image")
  v8i g1;
  g1[0] = (int)((2u << 16)      // data_size = 4 bytes
                | (1u << 20)    // pad_enable
                | (4u << 22));  // pad_interval code 4 = every 32 DWORDs
                                // pad_amount = 0 -> 1 DWORD
  g1[1] = (int)((d0 & 0xffffu) << 16);                             // dim0 lo
  g1[2] = (int)(((d0 >> 16) & 0xffffu) | ((d1 & 0xffffu) << 16));  // dim0 hi|dim1 lo
  g1[3] = (int)(((d1 >> 16) & 0xffffu) | (32u << 16));             // tile_dim0=32
  g1[4] = (int)32;                                                 // tile_dim1=32
  g1[5] = (int)(st & 0xffffffffu);      // tensor_dim0_stride[31:0]
  g1[6] = (int)((st >> 32) & 0xffffu);  // stride[47:32], dim1_stride unused
  g1[7] = 0;
  v4i z4 = {0, 0, 0, 0};
#if defined(__clang_major__) && (__clang_major__ >= 23)
  v8i z8 = {0, 0, 0, 0, 0, 0, 0, 0};
  __builtin_amdgcn_tensor_load_to_lds(g0, g1, z4, z4, z8, 0);
#else
  __builtin_amdgcn_tensor_load_to_lds(g0, g1, z4, z4, 0);
#endif
}

// ---------------------------------------------------------------------------
// Batched complex GEMM (planar re/im), C[bz] = A[bz] * (CB? conj(B) : B)[bz].
// Variants are compile-time: HA = A has imag plane, HB = B has imag plane.
// 32x32 output tile per block (128 threads = 4 waves, one 16x16 quadrant per
// wave), K tiled by 32. Global->LDS staging is done by the Tensor Data Mover,
// double-buffered: wave 0 issues tile t+1 then waits TENSORcnt<=NL so the DMA
// of the next tile overlaps this tile's WMMAs; barrier publishes to all waves.
// ---------------------------------------------------------------------------
#define TILE_PLANE_FLOATS (32 * 33)
#define TILE_PLANE_BYTES (TILE_PLANE_FLOATS * 4)

template <int HA, int HB, int CB>
__global__ __launch_bounds__(128) void k_cgemm_wmma(
    const float* __restrict__ Ar, const float* __restrict__ Ai,
    const float* __restrict__ Br, const float* __restrict__ Bi,
    float* __restrict__ Cr, float* __restrict__ Ci, int M, int N, int K,
    int lda, int ldb, int ldc, long long sA, long long sB, long long sC) {
  __shared__ float lds[2][4][TILE_PLANE_FLOATS];

  const long long bz = blockIdx.z;
  Ar += bz * sA;
  Br += bz * sB;
  Cr += bz * sC;
  Ci += bz * sC;
  if constexpr (HA) Ai += bz * sA;
  if constexpr (HB) Bi += bz * sB;

  const int tid = threadIdx.x;
  const int lane = tid & 31;
  const int wave = tid >> 5;
  const int half = lane >> 4;
  const int row = lane & 15;
  const int wm = (wave >> 1) * 16;
  const int wn = (wave & 1) * 16;
  const int m0 = blockIdx.y * 32;
  const int n0 = blockIdx.x * 32;

  const unsigned ldsBase = (unsigned)(uintptr_t)&lds[0][0][0];
  constexpr short NL = (short)(2 + HA + HB);  // TDM ops per tile

  auto issueTile = [&](int kt, int buf) {
    const int k0 = kt * 32;
    const unsigned b0 = ldsBase + (unsigned)buf * 4u * TILE_PLANE_BYTES;
    tdm_tile_f32(Ar + (long long)m0 * lda + k0, M - m0, K - k0, lda, b0);
    if constexpr (HA)
      tdm_tile_f32(Ai + (long long)m0 * lda + k0, M - m0, K - k0, lda,
                   b0 + TILE_PLANE_BYTES);
    tdm_tile_f32(Br + (long long)k0 * ldb + n0, K - k0, N - n0, ldb,
                 b0 + 2 * TILE_PLANE_BYTES);
    if constexpr (HB)
      tdm_tile_f32(Bi + (long long)k0 * ldb + n0, K - k0, N - n0, ldb,
                   b0 + 3 * TILE_PLANE_BYTES);
  };

  v8f rr = {}, ii = {}, ri = {}, ir = {};
  const int nT = (K + 31) >> 5;
  if (wave == 0) issueTile(0, 0);
  int cb = 0;
  for (int t = 0; t < nT; ++t) {
    if (wave == 0) {
      if (t + 1 < nT) {
        issueTile(t + 1, cb ^ 1);
        __builtin_amdgcn_s_wait_tensorcnt(NL);  // tile t finished
      } else {
        __builtin_amdgcn_s_wait_tensorcnt(0);
      }
    }
    __syncthreads();
    const float* Asr = lds[cb][0];
    const float* Asi = lds[cb][1];
    const float* Bsr = lds[cb][2];
    const float* Bsi = lds[cb][3];
#pragma unroll
    for (int kk = 0; kk < 32; kk += 4) {
      const int ka = kk + half * 2;
      v2f arf, aif, brf, bif;
      arf[0] = Asr[(wm + row) * 33 + ka];
      arf[1] = Asr[(wm + row) * 33 + ka + 1];
      brf[0] = Bsr[ka * 33 + wn + row];
      brf[1] = Bsr[(ka + 1) * 33 + wn + row];
      rr = wmma_f32(arf, brf, rr);  // Ar*Br
      if constexpr (HB) {
        bif[0] = Bsi[ka * 33 + wn + row];
        bif[1] = Bsi[(ka + 1) * 33 + wn + row];
        ri = wmma_f32(arf, bif, ri);  // Ar*Bi
      }
      if constexpr (HA) {
        aif[0] = Asi[(wm + row) * 33 + ka];
        aif[1] = Asi[(wm + row) * 33 + ka + 1];
        ir = wmma_f32(aif, brf, ir);  // Ai*Br
      }
      if constexpr (HA && HB) ii = wmma_f32(aif, bif, ii);  // Ai*Bi
    }
    __syncthreads();
    cb ^= 1;
  }

  // C layout: VGPR v -> M = v (lanes 0-15) / v+8 (lanes 16-31), N = lane%16
#pragma unroll
  for (int v = 0; v < 8; ++v) {
    int gm = m0 + wm + v + 8 * half;
    int gn = n0 + wn + row;
    if (gm < M && gn < N) {
      float cr, ci;
      if constexpr (HA && HB) {
        cr = CB ? (rr[v] + ii[v]) : (rr[v] - ii[v]);
        ci = CB ? (ir[v] - ri[v]) : (ir[v] + ri[v]);
      } else if constexpr (HA) {
        cr = rr[v];
        ci = ir[v];
      } else if constexpr (HB) {
        cr = rr[v];
        ci = CB ? -ri[v] : ri[v];
      } else {
        cr = rr[v];
        ci = 0.f;
      }
      long long o = (long long)gm * ldc + gn;
      Cr[o] = cr;
      Ci[o] = ci;
    }
  }
}

// ---------------------------------------------------------------------------
// Generic 5D gather-permute: out row-major over (d0..d4, d4 fastest),
// in-strides given per OUT dim. planes = 1 (real) or 2 (planar complex).
// ---------------------------------------------------------------------------
__global__ void k_permute5(const float* __restrict__ in, float* __restrict__ out,
                           long long total, int d1, int d2, int d3, int d4,
                           long long s0, long long s1, long long s2,
                           long long s3, long long s4, long long inPlane,
                           long long outPlane, int planes) {
  long long i = (long long)blockIdx.x * blockDim.x + threadIdx.x;
  if (i >= total) return;
  long long r = i;
  int i4 = (int)(r % d4); r /= d4;
  int i3 = (int)(r % d3); r /= d3;
  int i2 = (int)(r % d2); r /= d2;
  int i1 = (int)(r % d1); r /= d1;
  long long src = r * s0 + (long long)i1 * s1 + (long long)i2 * s2 +
                  (long long)i3 * s3 + (long long)i4 * s4;
  for (int p = 0; p < planes; ++p) out[p * outPlane + i] = in[p * inPlane + src];
}

// ---------------------------------------------------------------------------
// Constant builders (deterministic, rebuilt each launch; capture-safe)
// ---------------------------------------------------------------------------
__device__ __forceinline__ float sqrt_binom(int n, int k) {
  return expf(0.5f * (lgammaf((float)n + 1.f) - lgammaf((float)k + 1.f) -
                      lgammaf((float)(n - k) + 1.f)));
}
__device__ __forceinline__ float ipowf(float b, int e) {
  float r = 1.f;
  for (int i = 0; i < e; ++i) r *= b;
  return r;
}

// Wigner-d tables d^l_{mn}(beta): D[l][t][mi][ni], padded to Mdim=2L-1,
// via upward l-recurrence seeded with the closed-form boundary terms.
// mode 0: Driscoll-Healy betas for bandwidth bq (T nodes); mode 1: beta=pi/2.
__global__ void k_build_wigner(float* __restrict__ D, int L, int Mdim, int T,
                               int bq, int mode, long long total) {
  long long idx = (long long)blockIdx.x * blockDim.x + threadIdx.x;
  if (idx >= total) return;
  int ni = (int)(idx % Mdim);
  int mi = (int)((idx / Mdim) % Mdim);
  int t = (int)(idx / ((long long)Mdim * Mdim));
  int m = mi - (L - 1), n = ni - (L - 1);
  int am = m < 0 ? -m : m, an = n < 0 ? -n : n;
  int l0 = am > an ? am : an;
  float beta = mode ? 0.5f * PI_F : PI_F * (2 * t + 1) / (4.f * bq);
  long long strideL = (long long)T * Mdim * Mdim;
  float* base = D + (long long)t * Mdim * Mdim + (long long)mi * Mdim + ni;
  int lz = l0 < L ? l0 : L;
  for (int l = 0; l < lz; ++l) base[(long long)l * strideL] = 0.f;
  if (l0 >= L) return;

  float cb2 = cosf(0.5f * beta), sb2 = sinf(0.5f * beta);
  float dl;
  if (am >= an) {
    if (m == l0)
      dl = sqrt_binom(2 * l0, l0 - n) * ipowf(cb2, l0 + n) * ipowf(-sb2, l0 - n);
    else
      dl = sqrt_binom(2 * l0, l0 + n) * ipowf(cb2, l0 - n) * ipowf(sb2, l0 + n);
  } else {
    if (n == l0)
      dl = sqrt_binom(2 * l0, l0 - m) * ipowf(cb2, l0 + m) * ipowf(sb2, l0 - m);
    else
      dl = sqrt_binom(2 * l0, l0 + m) * ipowf(cb2, l0 - m) * ipowf(-sb2, l0 + m);
  }
  float dlm1 = 0.f;
  float x = cosf(beta);
  base[(long long)l0 * strideL] = dl;
  for (int l = l0; l < L - 1; ++l) {
    float lf = (float)l;
    float dn;
    if (l == 0) {
      dn = x * dl;
    } else {
      float c1 = (2.f * lf + 1.f) * ((lf + 1.f) * lf * x - (float)(m * n));
      float c2 = (lf + 1.f) * sqrtf((float)((l * l - m * m) * (l * l - n * n)));
      float c3 = lf * sqrtf((float)(((l + 1) * (l + 1) - m * m) *
                                    ((l + 1) * (l + 1) - n * n)));
      dn = (c1 * dl - c2 * dlm1) / c3;
    }
    dlm1 = dl;
    dl = dn;
    base[(long long)(l + 1) * strideL] = dl;
  }
}

// Driscoll-Healy quadrature weights, 2b nodes
__global__ void k_build_quad(float* __restrict__ w, int b) {
  int k = blockIdx.x * blockDim.x + threadIdx.x;
  if (k >= 2 * b) return;
  float beta = PI_F * (2 * k + 1) / (4.f * b);
  float s = 0.f;
  for (int j = 0; j < b; ++j) s += sinf((2 * j + 1) * beta) / (float)(2 * j + 1);
  w[k] = (2.f / b) * sinf(beta) * s;
}

// Forward DFT with idx_in gather folded in:  F[a][m] = exp(-2pi i a*idx/T)
// plus its transpose FT[m][a].
__global__ void k_build_fwd_dft(float* __restrict__ Fr, float* __restrict__ Fi,
                                float* __restrict__ FTr, float* __restrict__ FTi,
                                int T, int Mdim, int L, long long total) {
  long long idx = (long long)blockIdx.x * blockDim.x + threadIdx.x;
  if (idx >= total) return;
  int m = (int)(idx % Mdim);
  int a = (int)(idx / Mdim);
  int mv = m - (L - 1);
  int im = ((mv % T) + T) % T;
  float ang = -2.f * PI_F * (float)a * (float)im / (float)T;
  float c = cosf(ang), s = sinf(ang);
  Fr[(long long)a * Mdim + m] = c;
  Fi[(long long)a * Mdim + m] = s;
  FTr[(long long)m * T + a] = c;
  FTi[(long long)m * T + a] = s;
}

// Inverse DFT (scaled by n, idx_out scatter folded in):
// G[m][p] = exp(+2pi i io[m]*p/n), plus transpose GT[p][m].
__global__ void k_build_inv_dft(float* __restrict__ Gr, float* __restrict__ Gi,
                                float* __restrict__ GTr, float* __restrict__ GTi,
                                int n, int Mdim, int L, long long total) {
  long long idx = (long long)blockIdx.x * blockDim.x + threadIdx.x;
  if (idx >= total) return;
  int p = (int)(idx % n);
  int m = (int)(idx / n);
  int mv = m - (L - 1);
  int io = ((mv % n) + n) % n;
  float ang = 2.f * PI_F * (float)io * (float)p / (float)n;
  float c = cosf(ang), s = sinf(ang);
  Gr[(long long)m * n + p] = c;
  Gi[(long long)m * n + p] = s;
  GTr[(long long)p * Mdim + m] = c;
  GTi[(long long)p * Mdim + m] = s;
}

// A-tables (analysis): s2 layout [m][t][l]; so3 layout [m][n][t][l]
__global__ void k_build_Atab_s2(const float* __restrict__ D,
                                const float* __restrict__ w, float* __restrict__ A,
                                int L, int Mdim, int T, float anorm,
                                long long total) {
  long long idx = (long long)blockIdx.x * blockDim.x + threadIdx.x;
  if (idx >= total) return;
  int l = (int)(idx % L);
  int t = (int)((idx / L) % T);
  int m = (int)(idx / ((long long)L * T));
  A[idx] = anorm * w[t] *
           D[(((long long)l * T + t) * Mdim + m) * Mdim + (L - 1)];
}
__global__ void k_build_Atab_so3(const float* __restrict__ D,
                                 const float* __restrict__ w,
                                 float* __restrict__ A, int L, int Mdim, int T,
                                 float an2, long long total) {
  long long idx = (long long)blockIdx.x * blockDim.x + threadIdx.x;
  if (idx >= total) return;
  int l = (int)(idx % L);
  int t = (int)((idx / L) % T);
  int n = (int)((idx / ((long long)L * T)) % Mdim);
  int m = (int)(idx / ((long long)L * T * Mdim));
  A[idx] = an2 * w[t] * D[(((long long)l * T + t) * Mdim + m) * Mdim + n];
}

// K-tables: Dg(beta=pi/2) times phase exp(-i mv[m] alpha_g)
__global__ void k_build_Ktab_s2(const float* __restrict__ Dg,
                                float* __restrict__ Kr, float* __restrict__ Ki,
                                int L, int Mdim, int G, long long total) {
  long long idx = (long long)blockIdx.x * blockDim.x + threadIdx.x;
  if (idx >= total) return;
  int g = (int)(idx % G);
  int m = (int)((idx / G) % Mdim);
  int l = (int)(idx / ((long long)G * Mdim));
  float d = Dg[((long long)l * Mdim + m) * Mdim + (L - 1)];
  float a = 2.f * PI_F * (float)g / (float)G * (float)(m - (L - 1));
  Kr[idx] = d * cosf(a);
  Ki[idx] = -d * sinf(a);
}
__global__ void k_build_Ktab_so3(const float* __restrict__ Dg,
                                 float* __restrict__ Kr, float* __restrict__ Ki,
                                 int L, int Mdim, int G, long long total) {
  long long idx = (long long)blockIdx.x * blockDim.x + threadIdx.x;
  if (idx >= total) return;
  int g = (int)(idx % G);
  int n = (int)((idx / G) % Mdim);
  int m = (int)((idx / ((long long)G * Mdim)) % Mdim);
  int l = (int)(idx / ((long long)G * Mdim * Mdim));
  float d = Dg[((long long)l * Mdim + m) * Mdim + n];
  float a = 2.f * PI_F * (float)g / (float)G * (float)(m - (L - 1));
  Kr[idx] = d * cosf(a);
  Ki[idx] = -d * sinf(a);
}

// Synthesis table: S[m][n][l][t] = (2l+1) * d^l(beta_out)[t,m,n]
__global__ void k_build_Stab(const float* __restrict__ Do, float* __restrict__ S,
                             int L, int Mdim, int To, long long total) {
  long long idx = (long long)blockIdx.x * blockDim.x + threadIdx.x;
  if (idx >= total) return;
  int t = (int)(idx % To);
  int l = (int)((idx / To) % L);
  int n = (int)((idx / ((long long)To * L)) % Mdim);
  int m = (int)(idx / ((long long)To * L * Mdim));
  S[idx] =
      (float)(2 * l + 1) * Do[(((long long)l * To + t) * Mdim + m) * Mdim + n];
}

__global__ void k_build_wint(float* __restrict__ wt, int b) {
  int t = threadIdx.x;
  int T = 2 * b;
  if (t >= T) return;
  float beta = PI_F * (2 * t + 1) / (4.f * b);
  float s = 0.f;
  for (int j = 0; j < b; ++j) s += sinf((2 * j + 1) * beta) / (float)(2 * j + 1);
  float w = (2.f / b) * sinf(beta) * s;
  float a = PI_F / (float)b;  // 2pi/(2b)
  wt[t] = w * a * a / (8.f * PI_F * PI_F);
}

// ---------------------------------------------------------------------------
// BatchNorm (axes 0,2,3,4) stats + fused apply/ReLU
// ---------------------------------------------------------------------------
__global__ __launch_bounds__(256) void k_bn_stats(const float* __restrict__ x,
                                                  float* __restrict__ mv, int C,
                                                  int B, long long rest) {
  __shared__ float rs[256], rq[256];
  int c = blockIdx.x;
  float s = 0.f, q = 0.f;
  for (int b = 0; b < B; ++b) {
    const float* p = x + ((long long)b * C + c) * rest;
    for (long long i = threadIdx.x; i < rest; i += 256) {
      float v = p[i];
      s += v;
      q += v * v;
    }
  }
  rs[threadIdx.x] = s;
  rq[threadIdx.x] = q;
  __syncthreads();
  for (int st = 128; st > 0; st >>= 1) {
    if (threadIdx.x < st) {
      rs[threadIdx.x] += rs[threadIdx.x + st];
      rq[threadIdx.x] += rq[threadIdx.x + st];
    }
    __syncthreads();
  }
  if (threadIdx.x == 0) {
    float inv = 1.f / ((float)B * (float)rest);
    float m = rs[0] * inv;
    mv[2 * c] = m;
    mv[2 * c + 1] = rq[0] * inv - m * m;
  }
}

__global__ void k_bn_relu(const float* __restrict__ x, float* __restrict__ y,
                          const float* __restrict__ mv,
                          const float* __restrict__ g,
                          const float* __restrict__ be, int C, long long rest,
                          long long total) {
  long long i = (long long)blockIdx.x * blockDim.x + threadIdx.x;
  if (i >= total) return;
  int c = (int)((i / rest) % C);
  float m = mv[2 * c], v = mv[2 * c + 1];
  float r = (x[i] - m) * rsqrtf(v + BN_EPS) * g[c] + be[c];
  y[i] = fmaxf(r, 0.f);
}

// ---------------------------------------------------------------------------
// SO(3) integrate: out[b,c] = sum_t W_INT[t] * sum_{p,q} x[b,c,t,p,q]
// ---------------------------------------------------------------------------
__global__ __launch_bounds__(64) void k_integrate(const float* __restrict__ x,
                                                  const float* __restrict__ wt,
                                                  float* __restrict__ out, int T,
                                                  int nn) {
  __shared__ float red[64];
  int bc = blockIdx.x;
  const float* p = x + (long long)bc * T * nn;
  float acc = 0.f;
  for (int i = threadIdx.x; i < T * nn; i += 64) acc += p[i] * wt[i / nn];
  red[threadIdx.x] = acc;
  __syncthreads();
  for (int s = 32; s > 0; s >>= 1) {
    if (threadIdx.x < s) red[threadIdx.x] += red[threadIdx.x + s];
    __syncthreads();
  }
  if (threadIdx.x == 0) out[bc] = red[0];
}

// ---------------------------------------------------------------------------
// FC head: 5x (Linear + BN(batch) + ReLU) + Linear + sigmoid. Batch=4 fits in
// one workgroup; per-feature BN stats are thread-local (stats over 4 rows).
// ---------------------------------------------------------------------------
struct FCParams {
  const float* W[6];
  const float* b[6];
  const float* g[5];
  const float* be[5];
};

__global__ __launch_bounds__(256) void k_fc_head(const float* __restrict__ xin,
                                                 FCParams fp,
                                                 float* __restrict__ out) {
  __shared__ float buf[2][4][256];
  const int t = threadIdx.x;
  const int indim[6] = {256, 256, 190, 128, 64, 12};
  const int outdim[6] = {256, 190, 128, 64, 12, 1};
  for (int b = 0; b < 4; ++b) buf[0][b][t] = xin[b * 256 + t];
  __syncthreads();
  int cur = 0;
  for (int j = 0; j < 5; ++j) {
    if (t < outdim[j]) {
      float acc[4] = {0.f, 0.f, 0.f, 0.f};
      const float* Wr = fp.W[j] + (long long)t * indim[j];
      for (int k = 0; k < indim[j]; ++k) {
        float wv = Wr[k];
#pragma unroll
        for (int b = 0; b < 4; ++b) acc[b] += buf[cur][b][k] * wv;
      }
#pragma unroll
      for (int b = 0; b < 4; ++b) acc[b] += fp.b[j][t];
      float m = 0.25f * (acc[0] + acc[1] + acc[2] + acc[3]);
      float v = 0.f;
#pragma unroll
      for (int b = 0; b < 4; ++b) v += (acc[b] - m) * (acc[b] - m);
      v *= 0.25f;
      float sc = rsqrtf(v + BN_EPS) * fp.g[j][t];
#pragma unroll
      for (int b = 0; b < 4; ++b)
        buf[cur ^ 1][b][t] = fmaxf((acc[b] - m) * sc + fp.be[j][t], 0.f);
    } else {
      for (int b = 0; b < 4; ++b) buf[cur ^ 1][b][t] = 0.f;
    }
    __syncthreads();
    cur ^= 1;
  }
  if (t == 0) {
    for (int b = 0; b < 4; ++b) {
      float a = fp.b[5][0];
      for (int k = 0; k < 12; ++k) a += buf[cur][b][k] * fp.W[5][k];
      out[b] = 1.f / (1.f + expf(-a));
    }
  }
}

// ---------------------------------------------------------------------------
// Host orchestration
// ---------------------------------------------------------------------------
static const int BWv[7] = {32, 27, 22, 17, 12, 7, 2};
static const int CHv[7] = {6, 20, 40, 60, 120, 200, 256};

static inline unsigned gdiv(long long n, int t) {
  long long g = (n + t - 1) / t;
  return (unsigned)(g < 1 ? 1 : g);
}

static void cgemm(hipStream_t st, const float* Ar, const float* Ai,
                  const float* Br, const float* Bi, float* Cr, float* Ci, int M,
                  int N, int K, int lda, int ldb, int ldc, long long sA,
                  long long sB, long long sC, int batch, int conjB) {
  dim3 grid((N + 31) / 32, (M + 31) / 32, batch), blk(128);
  const bool ha = (Ai != nullptr), hb = (Bi != nullptr);
  if (ha && hb) {
    if (conjB)
      k_cgemm_wmma<1, 1, 1><<<grid, blk, 0, st>>>(Ar, Ai, Br, Bi, Cr, Ci, M, N,
                                                  K, lda, ldb, ldc, sA, sB, sC);
    else
      k_cgemm_wmma<1, 1, 0><<<grid, blk, 0, st>>>(Ar, Ai, Br, Bi, Cr, Ci, M, N,
                                                  K, lda, ldb, ldc, sA, sB, sC);
  } else if (ha) {
    k_cgemm_wmma<1, 0, 0><<<grid, blk, 0, st>>>(Ar, Ai, Br, Bi, Cr, Ci, M, N, K,
                                                lda, ldb, ldc, sA, sB, sC);
  } else if (hb) {
    k_cgemm_wmma<0, 1, 0><<<grid, blk, 0, st>>>(Ar, Ai, Br, Bi, Cr, Ci, M, N, K,
                                                lda, ldb, ldc, sA, sB, sC);
  } else {
    k_cgemm_wmma<0, 0, 0><<<grid, blk, 0, st>>>(Ar, Ai, Br, Bi, Cr, Ci, M, N, K,
                                                lda, ldb, ldc, sA, sB, sC);
  }
}

static void permute(hipStream_t st, const float* in, float* out, long long d0,
                    long long d1, long long d2, long long d3, long long d4,
                    long long s0, long long s1, long long s2, long long s3,
                    long long s4, long long inPlane, long long outPlane,
                    int planes) {
  long long total = d0 * d1 * d2 * d3 * d4;
  k_permute5<<<gdiv(total, 256), 256, 0, st>>>(in, out, total, (int)d1, (int)d2,
                                               (int)d3, (int)d4, s0, s1, s2, s3,
                                               s4, inPlane, outPlane, planes);
}

extern "C" void kernel_launch(void* const* d_in, const int* in_sizes, int n_in,
                              void* d_out, int out_size, void* d_ws,
                              size_t ws_size, hipStream_t stream) {
  (void)in_sizes;
  (void)n_in;
  (void)out_size;
  (void)ws_size;
  const int B = 4, G = 32;
  const float* x0 = (const float*)d_in[0];
  const float* wcp[6];
  const float* g3[6];
  const float* b3[6];
  for (int i = 0; i < 6; ++i) {
    wcp[i] = (const float*)d_in[1 + i];
    g3[i] = (const float*)d_in[7 + 2 * i];
    b3[i] = (const float*)d_in[8 + 2 * i];
  }
  FCParams fp;
  for (int j = 0; j < 6; ++j) {
    fp.W[j] = (const float*)d_in[19 + 2 * j];
    fp.b[j] = (const float*)d_in[20 + 2 * j];
  }
  for (int j = 0; j < 5; ++j) {
    fp.g[j] = (const float*)d_in[31 + 2 * j];
    fp.be[j] = (const float*)d_in[32 + 2 * j];
  }

  char* base = (char*)d_ws;
  size_t off = 0;
  auto alloc = [&](long long elems) -> float* {
    float* p = (float*)(base + off);
    off += (size_t)(((elems * 4 + 255) / 256) * 256);
    return p;
  };

  // persistent ping-pong activation buffers (max layer out = 13.7M floats)
  float* actA = alloc(14LL * 1000 * 1000);
  float* actB = alloc(14LL * 1000 * 1000);
  size_t scratchBase = off;

  const float* cur = x0;
  float* nxt = actA;

  for (int li = 0; li < 6; ++li) {
    off = scratchBase;  // reset scratch per layer
    const int b_in = BWv[li], b_out = BWv[li + 1];
    const int Ci = CHv[li], Co = CHv[li + 1];
    const int Ti = 2 * b_in, L = b_out, Mm = 2 * b_out - 1, To = 2 * b_out,
              no = 2 * b_out;
    const long long MM = (long long)Mm * Mm;

    // ---- constants -------------------------------------------------------
    float* w_in = alloc(Ti);
    k_build_quad<<<gdiv(Ti, 64), 64, 0, stream>>>(w_in, b_in);

    long long dInTot = (long long)Ti * Mm * Mm;
    float* Din = alloc((long long)L * dInTot);
    k_build_wigner<<<gdiv(dInTot, 128), 128, 0, stream>>>(Din, L, Mm, Ti, b_in,
                                                          0, dInTot);
    long long dgTot = MM;
    float* Dg = alloc((long long)L * MM);
    k_build_wigner<<<gdiv(dgTot, 128), 128, 0, stream>>>(Dg, L, Mm, 1, 0, 1,
                                                         dgTot);
    long long doTot = (long long)To * Mm * Mm;
    float* Dout = alloc((long long)L * doTot);
    k_build_wigner<<<gdiv(doTot, 128), 128, 0, stream>>>(Dout, L, Mm, To, b_out,
                                                         0, doTot);
    float anorm = PI_F / (float)b_in;
    float* Atab;
    if (li == 0) {
      long long n = (long long)Mm * Ti * L;
      Atab = alloc(n);
      k_build_Atab_s2<<<gdiv(n, 256), 256, 0, stream>>>(Din, w_in, Atab, L, Mm,
                                                        Ti, anorm, n);
    } else {
      long long n = MM * Ti * L;
      Atab = alloc(n);
      k_build_Atab_so3<<<gdiv(n, 256), 256, 0, stream>>>(
          Din, w_in, Atab, L, Mm, Ti, anorm * anorm, n);
    }
    long long kN = (li == 0) ? (long long)L * Mm * G : (long long)L * MM * G;
    float* Kt = alloc(2 * kN);
    if (li == 0)
      k_build_Ktab_s2<<<gdiv(kN, 256), 256, 0, stream>>>(Dg, Kt, Kt + kN, L, Mm,
                                                         G, kN);
    else
      k_build_Ktab_so3<<<gdiv(kN, 256), 256, 0, stream>>>(Dg, Kt, Kt + kN, L,
                                                          Mm, G, kN);
    long long sN = MM * L * To;
    float* St = alloc(sN);
    k_build_Stab<<<gdiv(sN, 256), 256, 0, stream>>>(Dout, St, L, Mm, To, sN);

    long long fN = (long long)Ti * Mm;
    float* Fin = alloc(2 * fN);
    float* FinT = alloc(2 * fN);
    k_build_fwd_dft<<<gdiv(fN, 256), 256, 0, stream>>>(
        Fin, Fin + fN, FinT, FinT + fN, Ti, Mm, L, fN);
    long long gN = (long long)Mm * no;
    float* Gi = alloc(2 * gN);
    float* GiT = alloc(2 * gN);
    k_build_inv_dft<<<gdiv(gN, 256), 256, 0, stream>>>(Gi, Gi + gN, GiT,
                                                       GiT + gN, no, Mm, L, gN);
    // pack conv weights [Ci,Co,G] -> wt[G][Ci*Co]
    float* wt = alloc((long long)G * Ci * Co);
    permute(stream, wcp[li], wt, 1, 1, G, Ci, Co, 0, 0, 1, (long long)Co * G, G,
            0, 0, 1);

    // ---- forward transform + per-degree GEMMs ----------------------------
    float* yh;
    const long long yhN = (long long)L * B * Mm * Mm * Co;
    if (li == 0) {
      // alpha DFT: xm[(b c t), m]
      long long R1 = (long long)B * Ci * Ti;
      float* xm = alloc(2 * R1 * Mm);
      cgemm(stream, cur, nullptr, Fin, Fin + fN, xm, xm + R1 * Mm, (int)R1, Mm,
            Ti, Ti, Mm, Mm, 0, 0, 0, 1, 0);
      // pack -> [m][(b c)][t]
      float* xmp = alloc(2 * R1 * Mm);
      permute(stream, xm, xmp, Mm, B, Ci, Ti, 1, 1, (long long)Ci * Ti * Mm,
              (long long)Ti * Mm, Mm, 0, R1 * Mm, R1 * Mm, 2);
      // xhat0[m][(b c)][l]  (contract t with Atab[m][t][l])
      long long BC = (long long)B * Ci;
      float* xhat0 = alloc(2 * Mm * BC * L);
      cgemm(stream, xmp, xmp + R1 * Mm, Atab, nullptr, xhat0,
            xhat0 + Mm * BC * L, (int)BC, L, Ti, Ti, L, L, BC * Ti,
            (long long)Ti * L, BC * L, Mm, 0);
      // pack -> xhA[l][b][m][i]
      long long xhAN = (long long)L * B * Mm * Ci;
      float* xhA = alloc(2 * xhAN);
      permute(stream, xhat0, xhA, L, B, Mm, Ci, 1, 1, (long long)Ci * L,
              BC * L, L, 0, Mm * BC * L, xhAN, 2);
      // khat0[(l n)][(i o)]
      long long khN = (long long)L * Mm * Ci * Co;
      float* kh0 = alloc(2 * khN);
      cgemm(stream, Kt, Kt + kN, wt, nullptr, kh0, kh0 + khN, L * Mm, Ci * Co,
            G, G, Ci * Co, Ci * Co, 0, 0, 0, 1, 0);
      // pack -> khB[l][i][n][o]
      float* khB = alloc(2 * khN);
      permute(stream, kh0, khB, L, Ci, Mm, Co, 1, (long long)Mm * Ci * Co,
              (long long)Co, (long long)Ci * Co, 1, 0, khN, khN, 2);
      // yh[l][(b m)][(n o)] = xhA * conj(khB), K = Ci
      yh = alloc(2 * yhN);
      cgemm(stream, xhA, xhA + xhAN, khB, khB + khN, yh, yh + yhN, B * Mm,
            Mm * Co, Ci, Ci, Mm * Co, Mm * Co, (long long)B * Mm * Ci,
            (long long)Ci * Mm * Co, (long long)B * Mm * Mm * Co, L, 1);
    } else {
      // gamma DFT
      long long R1 = (long long)B * Ci * Ti * Ti;
      float* x1 = alloc(2 * R1 * Mm);
      cgemm(stream, cur, nullptr, Fin, Fin + fN, x1, x1 + R1 * Mm, (int)R1, Mm,
            Ti, Ti, Mm, Mm, 0, 0, 0, 1, 0);
      // alpha DFT (batched, A = F^T)
      long long R1b = (long long)B * Ci * Ti;
      float* xmn = alloc(2 * R1b * MM);
      cgemm(stream, FinT, FinT + fN, x1, x1 + R1 * Mm, xmn, xmn + R1b * MM, Mm,
            Mm, Ti, Ti, Mm, Mm, 0, (long long)Ti * Mm, MM, (int)R1b, 0);
      // pack -> [m][n][(b c)][t]
      long long BC = (long long)B * Ci;
      float* xmp = alloc(2 * MM * BC * Ti);
      permute(stream, xmn, xmp, Mm, Mm, B, Ci, Ti, Mm, 1,
              (long long)Ci * Ti * MM, (long long)Ti * MM, MM, R1b * MM,
              MM * BC * Ti, 2);
      // xhat0[m][n][(b c)][l]
      float* xhat0 = alloc(2 * MM * BC * L);
      cgemm(stream, xmp, xmp + MM * BC * Ti, Atab, nullptr, xhat0,
            xhat0 + MM * BC * L, (int)BC, L, Ti, Ti, L, L, BC * Ti,
            (long long)Ti * L, BC * L, (int)MM, 0);
      // pack -> xhA[l][b][m][i][k]
      long long xhAN = (long long)L * B * Mm * Ci * Mm;
      float* xhA = alloc(2 * xhAN);
      permute(stream, xhat0, xhA, L, B, Mm, Ci, Mm, 1, (long long)Ci * L,
              (long long)Mm * BC * L, L, BC * L, MM * BC * L, xhAN, 2);
      // khat0[(l k n)][(i o)]
      long long khN = (long long)L * MM * Ci * Co;
      float* kh0 = alloc(2 * khN);
      cgemm(stream, Kt, Kt + kN, wt, nullptr, kh0, kh0 + khN,
            (int)((long long)L * MM), Ci * Co, G, G, Ci * Co, Ci * Co, 0, 0, 0,
            1, 0);
      // pack -> khB[l][i][k][n][o]
      float* khB = alloc(2 * khN);
      permute(stream, kh0, khB, L, Ci, Mm, Mm, Co, MM * Ci * Co, (long long)Co,
              (long long)Mm * Ci * Co, (long long)Ci * Co, 1, khN, khN, 2);
      // yh[l][(b m)][(n o)], K = Ci*Mm, conj(B)
      yh = alloc(2 * yhN);
      cgemm(stream, xhA, xhA + xhAN, khB, khB + khN, yh, yh + yhN, B * Mm,
            Mm * Co, Ci * Mm, Ci * Mm, Mm * Co, Mm * Co,
            (long long)B * Mm * Ci * Mm, (long long)Ci * Mm * Mm * Co,
            (long long)B * Mm * Mm * Co, L, 1);
    }

    // ---- synthesis (common) ---------------------------------------------
    // pack yh[l][b][m][n][o] -> yS[m][n][(b o)][l]
    long long ySN = MM * B * Co * L;
    float* yS = alloc(2 * ySN);
    permute(stream, yh, yS, Mm, Mm, B, Co, L, (long long)Mm * Co, (long long)Co,
            MM * Co, 1, (long long)B * MM * Co, yhN, ySN, 2);
    // s0[m][n][(b o)][t] = yS * Stab[m][n][l][t]
    long long s0N = MM * B * Co * To;
    float* s0 = alloc(2 * s0N);
    cgemm(stream, yS, yS + ySN, St, nullptr, s0, s0 + s0N, B * Co, To, L, L, To,
          To, (long long)B * Co * L, (long long)L * To, (long long)B * Co * To,
          (int)MM, 0);
    // pack -> sP[b][o][t][m][n]
    float* sP = alloc(2 * s0N);
    permute(stream, s0, sP, B, Co, To, Mm, Mm, (long long)Co * To,
            (long long)To, 1, Mm * (long long)B * Co * To,
            (long long)B * Co * To, s0N, s0N, 2);
    // inverse DFT along n then m (scaled IDFT; real part is the activation)
    long long R2 = (long long)B * Co * To * Mm;
    float* t1 = alloc(2 * R2 * no);
    cgemm(stream, sP, sP + s0N, Gi, Gi + gN, t1, t1 + R2 * no, (int)R2, no, Mm,
          Mm, no, no, 0, 0, 0, 1, 0);
    long long FNc = (long long)B * Co * To * no * no;
    float* Fo = alloc(2 * FNc);
    cgemm(stream, GiT, GiT + gN, t1, t1 + R2 * no, Fo, Fo + FNc, no, no, Mm, Mm,
          no, no, 0, (long long)Mm * no, (long long)no * no, B * Co * To, 0);

    // ---- BN + ReLU -------------------------------------------------------
    float* mv = alloc(2LL * Co);
    long long rest = (long long)To * no * no;
    k_bn_stats<<<Co, 256, 0, stream>>>(Fo, mv, Co, B, rest);
    float* dst = nxt;
    k_bn_relu<<<gdiv(FNc, 256), 256, 0, stream>>>(Fo, dst, mv, g3[li], b3[li],
                                                  Co, rest, FNc);
    cur = dst;
    nxt = (dst == actA) ? actB : actA;
  }

  // ---- integrate + FC head ----------------------------------------------
  float* wint = alloc(8);
  k_build_wint<<<1, 32, 0, stream>>>(wint, BWv[6]);
  float* feat = alloc((long long)B * 256);
  k_integrate<<<B * 256, 64, 0, stream>>>(cur, wint, feat, 2 * BWv[6],
                                          (2 * BWv[6]) * (2 * BWv[6]));
  k_fc_head<<<1, 256, 0, stream>>>(feat, fp, (float*)d_out);
}